// GCNModel_56865366999234
// MI455X (gfx1250) — compile-verified
//
#include <hip/hip_runtime.h>

typedef __attribute__((ext_vector_type(2))) float v2f;
typedef __attribute__((ext_vector_type(8))) float v8f;

__device__ __forceinline__ float atomic_fadd(float* p, float v) {
  // hardware global_atomic_add_f32 (coarse-grained device memory only)
  return unsafeAtomicAdd(p, v);
}

// ---- degree / normalization -------------------------------------------------

__global__ void deg_init_kernel(float* __restrict__ deg, int n) {
  int i = blockIdx.x * blockDim.x + threadIdx.x;
  if (i < n) deg[i] = 1.0f;  // self-loop contributes 1 to every node's degree
}

__global__ void deg_edge_kernel(const int* __restrict__ dst, float* __restrict__ deg, int e) {
  int i = blockIdx.x * blockDim.x + threadIdx.x;
  if (i < e) atomic_fadd(&deg[dst[i]], 1.0f);
}

__global__ void dinv_kernel(float* __restrict__ deg, int n) {
  int i = blockIdx.x * blockDim.x + threadIdx.x;
  if (i < n) {
    float d = deg[i];
    deg[i] = (d > 0.0f) ? rsqrtf(d) : 0.0f;  // in-place: deg -> dinv
  }
}

// ---- fp32 WMMA GEMM: xs = acc = (A @ B) * dinv[row] -------------------------
// One wave per 16x16 output tile. K stepped by 4 using V_WMMA_F32_16X16X4_F32.
// A-matrix f32 16x4 layout: lanes 0-15 rows M=lane (K=0,1), lanes 16-31 same M (K=2,3).
// B-matrix f32 4x16: lanes hold N; lane-half h, component j holds K = 2h+j.
// C/D 16x16 f32: VGPR r -> row = 8*half + r, col = lane&15.
// K and NCOLS are compile-time so every A/B access folds to base + imm offset.
template <int K, int NCOLS>
__global__ void gemm_scale_kernel(const float* __restrict__ A, const float* __restrict__ B,
                                  const float* __restrict__ dinv,
                                  float* __restrict__ xs, float* __restrict__ acc,
                                  int nrows) {
  const int lane = threadIdx.x & 31;
  const int wave = threadIdx.x >> 5;     // column tile handled by this wave
  const int half = lane >> 4;
  const int l16  = lane & 15;
  const int m0   = blockIdx.x << 4;
  const int ncol = (wave << 4) + l16;

  int mload = m0 + l16;
  if (mload >= nrows) mload = nrows - 1;  // clamp (stores are guarded below)
  const float* arow = A + (size_t)mload * K + (half << 1);
  const float* bcol = B + ncol;           // column slice; rows at imm offsets

  v8f c = {};
#pragma unroll
  for (int k0 = 0; k0 < K; k0 += 4) {
    const int kb = k0 + (half << 1);      // lane-half K offset for B
    v2f a;
    a.x = arow[k0];
    a.y = arow[k0 + 1];
    v2f b;
    b.x = bcol[kb * NCOLS];
    b.y = bcol[(kb + 1) * NCOLS];
    c = __builtin_amdgcn_wmma_f32_16x16x4_f32(false, a, false, b, (short)0, c,
                                              false, false);
  }

  const int rbase = m0 + (half << 3);
#pragma unroll
  for (int r = 0; r < 8; ++r) {
    const int row = rbase + r;
    if (row < nrows) {
      const float v = c[r] * dinv[row];   // pre-scale by dinv[src] once per node
      const size_t off = (size_t)row * NCOLS + ncol;
      xs[off]  = v;                        // read-only copy for edge gathers
      acc[off] = v;                        // accumulator seeded with self-loop term
    }
  }
}

// ---- edge scatter: acc[dst] += xs[src], 4 floats per thread ------------------
// NCOLS compile-time: chunk math becomes shifts/masks on constants.
template <int NCOLS>
__global__ void scatter_kernel(const int* __restrict__ src, const int* __restrict__ dst,
                               const float* __restrict__ xs, float* __restrict__ acc,
                               long long total) {
  constexpr int CPE = NCOLS / 4;          // 4-float chunks per edge
  long long t = (long long)blockIdx.x * blockDim.x + threadIdx.x;
  if (t >= total) return;
  const long long e = t / CPE;            // power of two -> shift
  const int ch = (int)(t % CPE);
  const int s = src[e];
  const int d = dst[e];
  const float4 v = *reinterpret_cast<const float4*>(xs + (size_t)s * NCOLS + (ch << 2));
  float* p = acc + (size_t)d * NCOLS + (ch << 2);
  atomic_fadd(p + 0, v.x);
  atomic_fadd(p + 1, v.y);
  atomic_fadd(p + 2, v.z);
  atomic_fadd(p + 3, v.w);
}

// ---- elementwise: buf = [relu](buf * dinv[row] + bias[col]) ------------------

template <int NCOLS, bool RELU>
__global__ void finish_kernel(float* __restrict__ buf, const float* __restrict__ dinv,
                              const float* __restrict__ bias, long long total) {
  long long t = (long long)blockIdx.x * blockDim.x + threadIdx.x;
  if (t >= total) return;
  const int row = (int)(t / NCOLS);
  const int col = (int)(t % NCOLS);
  float v = buf[t] * dinv[row] + bias[col];
  if (RELU) v = fmaxf(v, 0.0f);
  buf[t] = v;
}

// -----------------------------------------------------------------------------

extern "C" void kernel_launch(void* const* d_in, const int* in_sizes, int n_in,
                              void* d_out, int out_size, void* d_ws, size_t ws_size,
                              hipStream_t stream) {
  (void)n_in; (void)out_size; (void)ws_size;
  constexpr int IN = 128, HID = 64, CLS = 32;

  const float* x  = (const float*)d_in[0];
  const int*   ei = (const int*)d_in[1];
  // d_in[2] = edge_attr, unused by GCNConv
  const float* W1 = (const float*)d_in[3];
  const float* b1 = (const float*)d_in[4];
  const float* W2 = (const float*)d_in[5];
  const float* b2 = (const float*)d_in[6];
  float* out = (float*)d_out;

  const int n = in_sizes[0] / IN;   // 100000
  const int e = in_sizes[1] / 2;    // 1600000
  const int* src = ei;              // edge_index[0]
  const int* dst = ei + e;          // edge_index[1]

  // workspace layout (floats): dinv[n] | xs1[n*HID] | acc1[n*HID] (reused as h)
  float* ws   = (float*)d_ws;
  float* dinv = ws;
  float* xs1  = dinv + n;
  float* acc1 = xs1 + (size_t)n * HID;
  float* xs2  = xs1;                // layer-2 scaled features reuse xs1 buffer

  const int TB = 256;

  // normalization coefficients
  deg_init_kernel<<<(n + TB - 1) / TB, TB, 0, stream>>>(dinv, n);
  deg_edge_kernel<<<(e + TB - 1) / TB, TB, 0, stream>>>(dst, dinv, e);
  dinv_kernel<<<(n + TB - 1) / TB, TB, 0, stream>>>(dinv, n);

  const int mtiles = (n + 15) / 16;

  // layer 1: xs1 = acc1 = (x @ W1) * dinv   (WMMA f32 16x16x4, K fully unrolled)
  gemm_scale_kernel<IN, HID><<<mtiles, 32 * (HID / 16), 0, stream>>>(
      x, W1, dinv, xs1, acc1, n);
  // acc1[dst] += xs1[src] over all edges (16 chunks of 4 floats per edge)
  {
    long long tot = (long long)e * (HID / 4);
    scatter_kernel<HID><<<(unsigned)((tot + TB - 1) / TB), TB, 0, stream>>>(
        src, dst, xs1, acc1, tot);
  }
  // h = relu(acc1 * dinv + b1), in place
  {
    long long tot = (long long)n * HID;
    finish_kernel<HID, true><<<(unsigned)((tot + TB - 1) / TB), TB, 0, stream>>>(
        acc1, dinv, b1, tot);
  }

  // layer 2: xs2 = (h @ W2) * dinv, accumulator seeded directly in d_out
  gemm_scale_kernel<HID, CLS><<<mtiles, 32 * (CLS / 16), 0, stream>>>(
      acc1, W2, dinv, xs2, out, n);
  {
    long long tot = (long long)e * (CLS / 4);
    scatter_kernel<CLS><<<(unsigned)((tot + TB - 1) / TB), TB, 0, stream>>>(
        src, dst, xs2, out, tot);
  }
  // out = out * dinv + b2
  {
    long long tot = (long long)n * CLS;
    finish_kernel<CLS, false><<<(unsigned)((tot + TB - 1) / TB), TB, 0, stream>>>(
        out, dinv, b2, tot);
  }
}